// GNN_54494545052225
// MI455X (gfx1250) — compile-verified
//
#include <hip/hip_runtime.h>

typedef float v2f __attribute__((ext_vector_type(2)));
typedef float v8f __attribute__((ext_vector_type(8)));

#define N_NODES  100000
#define N_EDGES  1600000
#define HID      128
#define N_GRAPHS 512
#define BN_EPS   1e-5f

// ---------------------------------------------------------------- utilities
__global__ void fill_f32(float* __restrict__ p, float v, long n) {
  long i = (long)blockIdx.x * blockDim.x + threadIdx.x;
  if (i < n) p[i] = v;
}

// deg[dst] += 1 for every edge (deg pre-initialized to 1.0 = self loop)
__global__ void deg_accum(const int* __restrict__ ei, float* __restrict__ deg) {
  int e = blockIdx.x * blockDim.x + threadIdx.x;
  if (e < N_EDGES) atomicAdd(&deg[ei[N_EDGES + e]], 1.0f);
}

__global__ void dnorm_k(float* __restrict__ deg) {
  int i = blockIdx.x * blockDim.x + threadIdx.x;
  if (i < N_NODES) deg[i] = rsqrtf(deg[i]);   // deg >= 1 always (self loop)
}

// ---------------------------------------------------------------- WMMA GEMM
// C[M,Nc] = act(A[M,K] @ W[K,Nc] + bias). One block = 256 thr = 8 waves.
// blockIdx.x -> 16-row strip, blockIdx.y -> group of 128 cols, wave -> 16 cols.
// Uses exact-f32 V_WMMA_F32_16X16X4_F32.
template<bool RELU, bool BIAS>
__global__ void gemm16_wmma(const float* __restrict__ A, const float* __restrict__ W,
                            const float* __restrict__ bias, float* __restrict__ C,
                            int K, int Nc) {
  extern __shared__ float lds[];                    // 16 x K A-tile
  const int tid   = threadIdx.x;
  const int mBase = blockIdx.x * 16;
  const int nBase = blockIdx.y * 128 + (tid >> 5) * 16;

  for (int i = tid; i < 16 * K; i += 256)
    lds[i] = A[(size_t)(mBase + i / K) * K + (i % K)];
  __syncthreads();

  const int lane = tid & 31;
  const int l15  = lane & 15;
  const int klo  = (lane < 16) ? 0 : 2;             // A/B frag: hi half-wave holds K=2,3
  const float* ldsrow = lds + l15 * K;

  v8f acc = {};
  for (int k0 = 0; k0 < K; k0 += 4) {
    v2f a, b;
    a.x = ldsrow[k0 + klo];
    a.y = ldsrow[k0 + klo + 1];
    b.x = W[(size_t)(k0 + klo)     * Nc + nBase + l15];
    b.y = W[(size_t)(k0 + klo + 1) * Nc + nBase + l15];
    acc = __builtin_amdgcn_wmma_f32_16x16x4_f32(false, a, false, b,
                                                (short)0, acc, false, false);
  }

  const int col  = nBase + l15;
  const int row0 = mBase + ((lane < 16) ? 0 : 8);   // D layout: hi lanes hold M=8..15
  #pragma unroll
  for (int r = 0; r < 8; ++r) {
    float v = acc[r];
    if (BIAS) v += bias[col];
    if (RELU) v = fmaxf(v, 0.0f);
    C[(size_t)(row0 + r) * Nc + col] = v;
  }
}

// ---------------------------------------------------------------- GCN aggregation
// agg[i] = b + h[i]*dnorm[i]^2  (self loop + bias)
__global__ void init_agg(const float* __restrict__ h, const float* __restrict__ dnorm,
                         const float* __restrict__ b, float* __restrict__ agg) {
  long t = (long)blockIdx.x * blockDim.x + threadIdx.x;
  if (t >= (long)N_NODES * HID) return;
  int i = (int)(t >> 7), c = (int)(t & 127);
  float dn = dnorm[i];
  agg[t] = b[c] + h[t] * dn * dn;
}

// one edge per wave32; lane l handles channels 4l..4l+3
__global__ void edge_scatter(const int* __restrict__ ei, const float* __restrict__ h,
                             const float* __restrict__ dnorm, float* __restrict__ agg) {
  int wid  = (blockIdx.x * blockDim.x + threadIdx.x) >> 5;
  int lane = threadIdx.x & 31;
  if (wid >= N_EDGES) return;
  int s = ei[wid];
  int d = ei[N_EDGES + wid];
  float w = dnorm[s] * dnorm[d];
  float4 v = ((const float4*)(h + (size_t)s * HID))[lane];
  float* ad = agg + (size_t)d * HID + lane * 4;
  atomicAdd(ad + 0, v.x * w);
  atomicAdd(ad + 1, v.y * w);
  atomicAdd(ad + 2, v.z * w);
  atomicAdd(ad + 3, v.w * w);
}

// ---------------------------------------------------------------- BatchNorm + pool
// 128 threads/block (one per channel), 256 rows per block, coalesced row reads
__global__ void bn_stats(const float* __restrict__ agg, float* __restrict__ sums) {
  int c  = threadIdx.x;
  int r0 = blockIdx.x * 256;
  int r1 = min(r0 + 256, N_NODES);
  float s = 0.f, sq = 0.f;
  for (int r = r0; r < r1; ++r) {
    float v = agg[(size_t)r * HID + c];
    s += v; sq += v * v;
  }
  atomicAdd(&sums[c], s);
  atomicAdd(&sums[HID + c], sq);
}

__global__ void bn_finalize(const float* __restrict__ sums,
                            float* __restrict__ meanv, float* __restrict__ istd) {
  int c = threadIdx.x;
  float m   = sums[c] * (1.0f / N_NODES);
  float var = sums[HID + c] * (1.0f / N_NODES) - m * m;  // biased variance
  meanv[c] = m;
  istd[c]  = rsqrtf(var + BN_EPS);
}

__global__ void bn_pool(const float* __restrict__ agg, const float* __restrict__ meanv,
                        const float* __restrict__ istd, const float* __restrict__ gamma,
                        const float* __restrict__ beta, const int* __restrict__ batch,
                        float* __restrict__ pool, float* __restrict__ cnt) {
  int c = threadIdx.x;
  int r = blockIdx.x;
  float v = (agg[(size_t)r * HID + c] - meanv[c]) * istd[c] * gamma[c] + beta[c];
  int b = batch[r];
  atomicAdd(&pool[(size_t)b * HID + c], v);
  if (c == 0) atomicAdd(&cnt[b], 1.0f);
}

__global__ void pool_div(const float* __restrict__ pool, const float* __restrict__ cnt,
                         float* __restrict__ g) {
  int b = blockIdx.x, c = threadIdx.x;
  g[(size_t)b * HID + c] = pool[(size_t)b * HID + c] / fmaxf(cnt[b], 1.0f);
}

// ---------------------------------------------------------------- final 128->1
__global__ void final_layer(const float* __restrict__ a2, const float* __restrict__ W3,
                            const float* __restrict__ b3, float* __restrict__ out) {
  int g = blockIdx.x * blockDim.x + threadIdx.x;
  if (g >= N_GRAPHS) return;
  float s = b3[0];
  #pragma unroll 4
  for (int k = 0; k < HID; ++k) s += a2[(size_t)g * HID + k] * W3[k];
  out[g] = s;
}

// ---------------------------------------------------------------- driver
extern "C" void kernel_launch(void* const* d_in, const int* in_sizes, int n_in,
                              void* d_out, int out_size, void* d_ws, size_t ws_size,
                              hipStream_t stream) {
  const float* x     = (const float*)d_in[0];
  const int*   ei    = (const int*)  d_in[1];
  const int*   batch = (const int*)  d_in[2];
  const float* W_gcn = (const float*)d_in[3];
  const float* b_gcn = (const float*)d_in[4];
  const float* gamma = (const float*)d_in[5];
  const float* beta  = (const float*)d_in[6];
  const float* W1    = (const float*)d_in[7];
  const float* b1    = (const float*)d_in[8];
  const float* W2    = (const float*)d_in[9];
  const float* b2    = (const float*)d_in[10];
  const float* W3    = (const float*)d_in[11];
  const float* b3    = (const float*)d_in[12];
  float* out = (float*)d_out;

  float* ws = (float*)d_ws;
  size_t o = 0;
  float* hbuf  = ws + o; o += (size_t)N_NODES * HID;   // 12.8M
  float* agg   = ws + o; o += (size_t)N_NODES * HID;   // 12.8M
  float* deg   = ws + o; o += N_NODES;                 // becomes dnorm in place
  float* sums  = ws + o; o += 2 * HID;                 // sum, sumsq
  float* meanv = ws + o; o += HID;
  float* istd  = ws + o; o += HID;
  float* pool  = ws + o; o += (size_t)N_GRAPHS * HID;
  float* cnt   = ws + o; o += N_GRAPHS;
  float* gbuf  = ws + o; o += (size_t)N_GRAPHS * HID;
  float* a1    = ws + o; o += (size_t)N_GRAPHS * 256;
  float* a2    = ws + o; o += (size_t)N_GRAPHS * HID;

  // degree (init 1.0 for self loop), then accumulate, then rsqrt in place
  fill_f32<<<(N_NODES + 255) / 256, 256, 0, stream>>>(deg, 1.0f, N_NODES);
  // zero stats region: sums..cnt are contiguous (256+128+128+65536+512 floats)
  long statsN = 2 * HID + HID + HID + (long)N_GRAPHS * HID + N_GRAPHS;
  fill_f32<<<(int)((statsN + 255) / 256), 256, 0, stream>>>(sums, 0.0f, statsN);
  deg_accum<<<(N_EDGES + 255) / 256, 256, 0, stream>>>(ei, deg);
  dnorm_k<<<(N_NODES + 255) / 256, 256, 0, stream>>>(deg);

  // h = x @ W_gcn  (exact f32 WMMA; 6250 row strips)
  gemm16_wmma<false, false><<<dim3(N_NODES / 16, 1), 256, 16 * 128 * 4, stream>>>(
      x, W_gcn, nullptr, hbuf, 128, 128);

  // agg = b + self-loop, then scatter edges (one edge per wave32)
  init_agg<<<(int)(((long)N_NODES * HID + 255) / 256), 256, 0, stream>>>(hbuf, deg, b_gcn, agg);
  edge_scatter<<<N_EDGES / 8, 256, 0, stream>>>(ei, hbuf, deg, agg);

  // batchnorm (training stats) + mean pool
  bn_stats<<<(N_NODES + 255) / 256, 128, 0, stream>>>(agg, sums);
  bn_finalize<<<1, 128, 0, stream>>>(sums, meanv, istd);
  bn_pool<<<N_NODES, 128, 0, stream>>>(agg, meanv, istd, gamma, beta, batch, pool, cnt);
  pool_div<<<N_GRAPHS, 128, 0, stream>>>(pool, cnt, gbuf);

  // MLP: 512x128 @ 128x256 -> ReLU -> @ 256x128 -> ReLU -> @ 128x1
  gemm16_wmma<true, true><<<dim3(N_GRAPHS / 16, 2), 256, 16 * 128 * 4, stream>>>(
      gbuf, W1, b1, a1, 128, 256);
  gemm16_wmma<true, true><<<dim3(N_GRAPHS / 16, 1), 256, 16 * 256 * 4, stream>>>(
      a1, W2, b2, a2, 256, 128);
  final_layer<<<(N_GRAPHS + 255) / 256, 256, 0, stream>>>(a2, W3, b3, out);
}